// RecurrentSheafLayer_65652870087353
// MI455X (gfx1250) — compile-verified
//
#include <hip/hip_runtime.h>
#include <hip/hip_bf16.h>
#include <math.h>

typedef __attribute__((ext_vector_type(2))) float v2f;
typedef __attribute__((ext_vector_type(8))) float v8f;

#define D_DIM 1024
#define B_DIM 8
#define L_DIM 2048
#define NROWS (B_DIM * L_DIM)          // 16384
#define SCAN_WGS 32
#define ECOLS 32                        // e-columns per scan WG
#define WR_STRIDE 48                    // padded W_r row stride (dwords)
#define HS_STRIDE 1028                  // padded h row stride (dwords)

// ---------------------------------------------------------------------------
// f32 WMMA: D = A(16x4) * B(4x16) + C, exact fp32 on the matrix pipe.
// A: lane&15 = M row; VGPR0 = K + 2*(lane>>4), VGPR1 = K+1+2*(lane>>4)
// B: lane&15 = N col; same K striping.
// ---------------------------------------------------------------------------
__device__ __forceinline__ v8f wmma_f32(v2f a, v2f b, v8f c) {
    return __builtin_amdgcn_wmma_f32_16x16x4_f32(false, a, false, b,
                                                 (short)0, c, false, false);
}

// ---------------------------------------------------------------------------
// Grid barrier (sense-reversing) over SCAN_WGS persistent workgroups.
// ---------------------------------------------------------------------------
__device__ __forceinline__ void grid_sync(unsigned* bar, unsigned nwg) {
    __threadfence();
    __syncthreads();
    if (threadIdx.x == 0) {
        volatile unsigned* vgen = bar + 1;
        unsigned gen = *vgen;
        unsigned prev = atomicAdd(bar, 1u);
        if (prev + 1u == nwg) {
            *(volatile unsigned*)bar = 0u;
            __threadfence();
            atomicAdd(bar + 1, 1u);
        } else {
            while (*vgen == gen) { __builtin_amdgcn_s_sleep(2); }
        }
    }
    __syncthreads();
    __threadfence();
}

__device__ __forceinline__ float sigmoidf_dev(float v) {
    return 1.0f / (1.0f + expf(-v));
}

// ---------------------------------------------------------------------------
// Init: zero h ping-pong buffers and barrier counters (every call).
// ---------------------------------------------------------------------------
__global__ __launch_bounds__(256) void init_kernel(float* hbuf, unsigned* bar) {
    int i = blockIdx.x * 256 + threadIdx.x;
    if (i < 2 * B_DIM * D_DIM) hbuf[i] = 0.0f;
    if (i < 2) bar[i] = 0u;
}

// ---------------------------------------------------------------------------
// Tiled f32-WMMA GEMM: out[r,e] = act( sum_k A[r,k]*W[e,k] + bias[e] )
// MODE 0: sigmoid + per-WG gate partials.  MODE 1: identity.
// Tile: 64(M) x 32(N) per WG, 8 waves each own one 16x16 tile, Kc = 32.
// ---------------------------------------------------------------------------
template <int MODE>
__global__ __launch_bounds__(256)
void gemm_kernel(const float* __restrict__ A, const float* __restrict__ W,
                 const float* __restrict__ bias, float* __restrict__ out,
                 float* __restrict__ gate_partial) {
    __shared__ float As[64 * 34];   // stride 34: aligned b64 A pairs, no bank clash
    __shared__ float Bs[32 * 48];   // stride 48: lanes16-31 (+96 dw) hit disjoint banks
    __shared__ float redg[256];

    const int t = threadIdx.x;
    const int wave = t >> 5, lane = t & 31;
    const int mi = wave & 3, ni = wave >> 2;
    const int kh = lane >> 4, ln = lane & 15;
    const int wg = blockIdx.x;
    const int mb = wg >> 5, nb = wg & 31;
    const int r0 = mb * 64, e0 = nb * 32;

    v8f c = {0.f, 0.f, 0.f, 0.f, 0.f, 0.f, 0.f, 0.f};

    const int arow = t >> 2;            // 0..63
    const int acol = (t & 3) * 8;       // 0,8,16,24
    const int brow = t >> 3;            // 0..31 (e)
    const int bcol = (t & 7) * 4;       // 0..28 (k)

    for (int k0 = 0; k0 < D_DIM; k0 += 32) {
        __syncthreads();
        {   // Stage A tile (64x32)
            const float4* src =
                (const float4*)(A + (size_t)(r0 + arow) * D_DIM + k0 + acol);
            float4 v0 = src[0], v1 = src[1];
            float* dst = &As[arow * 34 + acol];
            dst[0] = v0.x; dst[1] = v0.y; dst[2] = v0.z; dst[3] = v0.w;
            dst[4] = v1.x; dst[5] = v1.y; dst[6] = v1.z; dst[7] = v1.w;
        }
        {   // Stage B tile transposed: Bs[k][n] = W[e0+n][k0+k]
            const float4* src =
                (const float4*)(W + (size_t)(e0 + brow) * D_DIM + k0 + bcol);
            float4 v = src[0];
            Bs[(bcol + 0) * 48 + brow] = v.x;
            Bs[(bcol + 1) * 48 + brow] = v.y;
            Bs[(bcol + 2) * 48 + brow] = v.z;
            Bs[(bcol + 3) * 48 + brow] = v.w;
        }
        __syncthreads();
        if (k0 + 32 < D_DIM) {  // -> global_prefetch_b8 on the streaming A
            __builtin_prefetch(A + (size_t)(r0 + arow) * D_DIM + k0 + 32 + acol, 0, 0);
        }
#pragma unroll
        for (int kk = 0; kk < 32; kk += 4) {
            const int ka = kk + 2 * kh;
            v2f a = *(const v2f*)(&As[(mi * 16 + ln) * 34 + ka]);
            v2f b;
            b.x = Bs[ka * 48 + ni * 16 + ln];
            b.y = Bs[(ka + 1) * 48 + ni * 16 + ln];
            c = wmma_f32(a, b, c);
        }
    }

    // Epilogue: VGPR i -> M = mi*16 + i + 8*kh, N = ni*16 + ln
    float gsum = 0.0f;
    const int e = e0 + ni * 16 + ln;
    const float bv = bias[e];
#pragma unroll
    for (int i = 0; i < 8; ++i) {
        const int m = r0 + mi * 16 + i + 8 * kh;
        float v = c[i] + bv;
        if (MODE == 0) { v = sigmoidf_dev(v); gsum += v; }
        out[(size_t)m * D_DIM + e] = v;
    }
    if (MODE == 0) {
        redg[t] = gsum;
        __syncthreads();
        for (int s = 128; s > 0; s >>= 1) {
            if (t < s) redg[t] += redg[t + s];
            __syncthreads();
        }
        if (t == 0) gate_partial[wg] = redg[0];
    }
}

// ---------------------------------------------------------------------------
// Persistent scan: 32 WGs x 32 e-columns. W_r slice (192 KB padded) + a
// 16-row zero-padded h staging buffer live in LDS (271 KB total — CDNA5's
// 320 KB/WGP). Per step: 8 waves = 2 e-tiles x 4 K-quarters, 64 chained
// f32 WMMAs each with a LITERAL trip count (scalar loop control, no exec
// masking), LDS-reduce the 4 K-quarter partials, elementwise recurrence,
// 32-WG grid barrier.
// ---------------------------------------------------------------------------
__global__ __launch_bounds__(256)
void scan_kernel(const float* __restrict__ x, const float* __restrict__ z,
                 const float* __restrict__ Wr, const float* __restrict__ br,
                 const float* __restrict__ decay, float* __restrict__ hbuf,
                 float* __restrict__ y, float* __restrict__ epart,
                 unsigned* __restrict__ bar) {
    extern __shared__ float smem[];
    float* wr  = smem;                        // [1024][WR_STRIDE]
    float* hs  = wr + D_DIM * WR_STRIDE;      // [16][HS_STRIDE], rows 8-15 = 0
    float* red = hs + 16 * HS_STRIDE;         // [8][256] wave partial C tiles
    float* red2 = red + 8 * 256;              // [128]

    const int t = threadIdx.x;
    const int wave = t >> 5, lane = t & 31;
    const int etile = wave & 1;               // which 16-col sub-tile
    const int kq = wave >> 1;                 // K quarter (256 wide)
    const int kh = lane >> 4, ln = lane & 15;
    const int wg = blockIdx.x;
    const int e0 = wg * ECOLS;

    // Stage W_r slice transposed (once): wr[k*48 + ec] = Wr[e0+ec][k]
    for (int idx = t; idx < ECOLS * D_DIM; idx += 256) {
        const int ec = idx >> 10, k = idx & (D_DIM - 1);
        wr[k * WR_STRIDE + ec] = Wr[(size_t)(e0 + ec) * D_DIM + k];
    }
    // Zero padded A rows 8..15 (once) so WMMA A-loads are unconditional.
    for (int idx = t; idx < 8 * HS_STRIDE; idx += 256)
        hs[8 * HS_STRIDE + idx] = 0.0f;

    // Per-thread constants for the elementwise phase (t<128: m=t>>4, n=t&15).
    const int m_el = t >> 4, n_el = t & 15;
    float ddv[2], brv[2];
#pragma unroll
    for (int j = 0; j < 2; ++j) {
        const int e = e0 + j * 16 + n_el;
        ddv[j] = sigmoidf_dev(decay[e]);
        brv[j] = br[e];
    }

    const int kbase = kq * 256;
    // Wave-constant base pointers; literal 64-iteration loop below.
    const float* ap_base = hs + ln * HS_STRIDE + 2 * kh + kbase;   // b64-aligned
    const float* bp_base = wr + etile * 16 + ln +
                           (size_t)(kbase + 2 * kh) * WR_STRIDE;

    for (int step = 0; step < L_DIM; ++step) {
        const float* hin = hbuf + (step & 1) * (B_DIM * D_DIM);
        float* hout = hbuf + ((step + 1) & 1) * (B_DIM * D_DIM);

        // Stage h (8x1024 = 32 KB) into LDS rows 0..7.
        for (int idx = t; idx < (B_DIM * D_DIM) / 4; idx += 256) {
            float4 v = ((const float4*)hin)[idx];
            const int row = idx >> 8, col = (idx & 255) << 2;
            float* dp = &hs[row * HS_STRIDE + col];
            dp[0] = v.x; dp[1] = v.y; dp[2] = v.z; dp[3] = v.w;
        }
        __syncthreads();

        // Prefetch next step's x/z (consumed by the elementwise phase on the
        // serial chain) while the WMMA pipe is busy.
        if (step + 1 < L_DIM && t < 128) {
#pragma unroll
            for (int j = 0; j < 2; ++j) {
                const size_t gn =
                    ((size_t)m_el * L_DIM + step + 1) * D_DIM + e0 + j * 16 + n_el;
                __builtin_prefetch(x + gn, 0, 0);
                __builtin_prefetch(z + gn, 0, 0);
            }
        }

        // pred partial = h[16(pad) x 256] * Wr^T[256 x 16]: 64 chained WMMAs,
        // literal trip count -> scalar loop control.
        v8f c = {0.f, 0.f, 0.f, 0.f, 0.f, 0.f, 0.f, 0.f};
#pragma unroll 8
        for (int it = 0; it < 64; ++it) {
            v2f a = *(const v2f*)(ap_base + it * 4);
            v2f b;
            b.x = bp_base[it * 4 * WR_STRIDE];
            b.y = bp_base[it * 4 * WR_STRIDE + WR_STRIDE];
            c = wmma_f32(a, b, c);
        }
        // Dump wave partial tiles: red[wave][M*16+N]
#pragma unroll
        for (int i = 0; i < 8; ++i)
            red[wave * 256 + (i + 8 * kh) * 16 + ln] = c[i];
        __syncthreads();

        // Combine 4 K-quarters + elementwise recurrence (threads 0..127).
        float sse = 0.0f;
        if (t < 128) {
#pragma unroll
            for (int j = 0; j < 2; ++j) {
                const int mn = m_el * 16 + n_el;
                const float pred = red[(j + 0) * 256 + mn] +
                                   red[(j + 2) * 256 + mn] +
                                   red[(j + 4) * 256 + mn] +
                                   red[(j + 6) * 256 + mn] + brv[j];
                const int e = e0 + j * 16 + n_el;
                const size_t g = ((size_t)m_el * L_DIM + step) * D_DIM + e;
                const float xt = x[g];
                const float raw = xt - pred;
                sse += raw * raw;
                const float err = tanhf(raw);
                const float zt = z[g];
                const float dd = ddv[j];
                const float hnew =
                    dd * hs[m_el * HS_STRIDE + e] + (1.0f - dd) * (zt * err);
                hout[m_el * D_DIM + e] = hnew;
                y[g] = hnew;
            }
            red2[t] = sse;
        }
        __syncthreads();
        if (t < B_DIM) {  // per-(step,batch) squared-norm partial, no atomics
            float s = 0.0f;
#pragma unroll
            for (int i = 0; i < 16; ++i) s += red2[t * 16 + i];
            epart[((size_t)wg * L_DIM + step) * B_DIM + t] = s;
        }
        grid_sync(bar, SCAN_WGS);
    }
}

// ---------------------------------------------------------------------------
// LayerNorm over D=1024 per row, in place on y.
// ---------------------------------------------------------------------------
__global__ __launch_bounds__(256)
void ln_kernel(float* __restrict__ y, const float* __restrict__ g,
               const float* __restrict__ b) {
    __shared__ float red[256];
    const int t = threadIdx.x;
    float* p = y + (size_t)blockIdx.x * D_DIM;
    float4 v = ((float4*)p)[t];
    red[t] = v.x + v.y + v.z + v.w;
    __syncthreads();
    for (int s = 128; s > 0; s >>= 1) {
        if (t < s) red[t] += red[t + s];
        __syncthreads();
    }
    const float mu = red[0] * (1.0f / D_DIM);
    __syncthreads();
    const float dx = v.x - mu, dy = v.y - mu, dz = v.z - mu, dw = v.w - mu;
    red[t] = dx * dx + dy * dy + dz * dz + dw * dw;
    __syncthreads();
    for (int s = 128; s > 0; s >>= 1) {
        if (t < s) red[t] += red[t + s];
        __syncthreads();
    }
    const float rs = rsqrtf(red[0] * (1.0f / D_DIM) + 1e-5f);
    float4 gv = ((const float4*)g)[t];
    float4 bv = ((const float4*)b)[t];
    float4 o;
    o.x = dx * rs * gv.x + bv.x;
    o.y = dy * rs * gv.y + bv.y;
    o.z = dz * rs * gv.z + bv.z;
    o.w = dw * rs * gv.w + bv.w;
    ((float4*)p)[t] = o;
}

// ---------------------------------------------------------------------------
// rho partials: ||W_r - I||_F^2 per-WG partials (deterministic).
// ---------------------------------------------------------------------------
__global__ __launch_bounds__(256)
void rho_kernel(const float* __restrict__ Wr, float* __restrict__ rhop) {
    __shared__ float red[256];
    const int t = threadIdx.x;
    const size_t base = (size_t)blockIdx.x * 16384;
    float s = 0.0f;
    for (size_t idx = base + t; idx < base + 16384; idx += 256) {
        const size_t r = idx >> 10, c = idx & 1023;
        const float v = Wr[idx] - ((r == c) ? 1.0f : 0.0f);
        s += v * v;
    }
    red[t] = s;
    __syncthreads();
    for (int k = 128; k > 0; k >>= 1) {
        if (t < k) red[t] += red[t + k];
        __syncthreads();
    }
    if (t == 0) rhop[blockIdx.x] = red[0];
}

// ---------------------------------------------------------------------------
// Finalize scalars in fixed order (deterministic): energy, gate, rho.
// ---------------------------------------------------------------------------
__global__ __launch_bounds__(256)
void finalize_kernel(const float* __restrict__ epart,
                     const float* __restrict__ gatep,
                     const float* __restrict__ rhop, float* __restrict__ outs) {
    __shared__ float red[256];
    const int t = threadIdx.x;

    float es = 0.0f;
    for (int sb = t; sb < L_DIM * B_DIM; sb += 256) {
        float s = 0.0f;
#pragma unroll
        for (int w = 0; w < SCAN_WGS; ++w)
            s += epart[(size_t)w * L_DIM * B_DIM + sb];
        es += sqrtf(s);
    }
    red[t] = es;
    __syncthreads();
    for (int k = 128; k > 0; k >>= 1) { if (t < k) red[t] += red[t + k]; __syncthreads(); }
    if (t == 0) outs[0] = red[0] / (float)(L_DIM * B_DIM);
    __syncthreads();

    float gs = 0.0f;
    for (int i = t; i < 8192; i += 256) gs += gatep[i];
    red[t] = gs;
    __syncthreads();
    for (int k = 128; k > 0; k >>= 1) { if (t < k) red[t] += red[t + k]; __syncthreads(); }
    if (t == 0) outs[1] = red[0] / ((float)NROWS * (float)D_DIM);
    __syncthreads();

    float rs = (t < 64) ? rhop[t] : 0.0f;
    red[t] = rs;
    __syncthreads();
    for (int k = 128; k > 0; k >>= 1) { if (t < k) red[t] += red[t + k]; __syncthreads(); }
    if (t == 0) outs[2] = sqrtf(red[0]);
}

// ---------------------------------------------------------------------------
extern "C" void kernel_launch(void* const* d_in, const int* in_sizes, int n_in,
                              void* d_out, int out_size, void* d_ws,
                              size_t ws_size, hipStream_t stream) {
    const float* x     = (const float*)d_in[0];
    const float* decay = (const float*)d_in[1];
    const float* W_r   = (const float*)d_in[2];
    const float* b_r   = (const float*)d_in[3];
    const float* W_u   = (const float*)d_in[4];
    const float* b_u   = (const float*)d_in[5];
    const float* W_o   = (const float*)d_in[6];
    const float* b_o   = (const float*)d_in[7];
    const float* ln_g  = (const float*)d_in[8];
    const float* ln_b  = (const float*)d_in[9];

    float* ws = (float*)d_ws;
    float* zbuf  = ws;                              // 16777216
    float* ybuf  = zbuf + (size_t)NROWS * D_DIM;    // 16777216
    float* hbuf  = ybuf + (size_t)NROWS * D_DIM;    // 16384 (ping-pong h)
    float* epart = hbuf + 2 * B_DIM * D_DIM;        // 32*2048*8 = 524288
    float* gatep = epart + (size_t)SCAN_WGS * L_DIM * B_DIM;  // 8192
    float* rhop  = gatep + 8192;                    // 64
    unsigned* bar = (unsigned*)(rhop + 64);         // 2

    float* out_f = (float*)d_out;
    float* scal  = out_f + (size_t)NROWS * D_DIM;

    const int scan_smem =
        (D_DIM * WR_STRIDE + 16 * HS_STRIDE + 8 * 256 + 128) * (int)sizeof(float);
    (void)hipFuncSetAttribute((const void*)scan_kernel,
                              hipFuncAttributeMaxDynamicSharedMemorySize,
                              scan_smem);

    init_kernel<<<64, 256, 0, stream>>>(hbuf, bar);
    gemm_kernel<0><<<(NROWS / 64) * (D_DIM / 32), 256, 0, stream>>>(
        x, W_u, b_u, zbuf, gatep);
    rho_kernel<<<64, 256, 0, stream>>>(W_r, rhop);
    scan_kernel<<<SCAN_WGS, 256, scan_smem, stream>>>(
        x, zbuf, W_r, b_r, decay, hbuf, ybuf, epart, bar);
    ln_kernel<<<NROWS, 256, 0, stream>>>(ybuf, ln_g, ln_b);
    gemm_kernel<1><<<(NROWS / 64) * (D_DIM / 32), 256, 0, stream>>>(
        ybuf, W_o, b_o, out_f, nullptr);
    finalize_kernel<<<1, 256, 0, stream>>>(epart, gatep, rhop, scal);
}